// S4Layer_12644383719491
// MI455X (gfx1250) — compile-verified
//
#include <hip/hip_runtime.h>
#include <math.h>
#include <stdint.h>

typedef float v2f __attribute__((ext_vector_type(2)));
typedef float v8f __attribute__((ext_vector_type(8)));

#define DM 512
#define NS 64
#define LL 2048
#define NB 16
#define T  16
#define NCHUNK (LL / T)

// workspace layout per channel d (floats)
#define OFF_K   0                 // 16 Toeplitz taps
#define OFF_G   16                // 16 x 128 state->output matrix
#define OFF_W   (16 + 2048)       // 128 x 16 injection matrix (row-permuted)
#define OFF_A16 (16 + 2048 + 2048)// 64 complex A^16 (re,im interleaved)
#define PD      4352              // padded per-d stride

// LDS tile: [s][b] pairs of (d0,d0+1), padded stride 17 to avoid bank conflicts
#define TP 17

// Row permutation so that the injection WMMA's C/D-layout output lands exactly
// in the B-operand fragment slots of the resident state (no cross-lane moves).
__device__ __forceinline__ int permM(int S) {
    return (S & ~15) + (((S >> 1) & 1) << 3) + (((S >> 2) & 3) << 1) + (S & 1);
}

__device__ __forceinline__ uint32_t lds_off(const void* p) {
    return (uint32_t)(uintptr_t)p;
}

// ---------------------------------------------------------------------------
// Kernel A: per-channel table precompute. One block per d, one thread per n.
// ---------------------------------------------------------------------------
__global__ void s4_tables(const float* __restrict__ lar, const float* __restrict__ aim,
                          const float* __restrict__ Bm,  const float* __restrict__ Cm,
                          const float* __restrict__ logdt, float* __restrict__ ws) {
    const int d = blockIdx.x;
    const int n = threadIdx.x;
    const float dt  = expf(logdt[0]);
    const int  idx  = d * NS + n;
    const float adr = 1.0f - expf(lar[idx]) * dt;   // Re(1 + A*dt)
    const float adi = aim[idx] * dt;                // Im(1 + A*dt)
    const float Bd  = Bm[idx] * dt;
    const float Cr  = Cm[idx * 2 + 0];
    const float Ci  = Cm[idx * 2 + 1];

    float pr[17], pi[17];
    pr[0] = 1.0f; pi[0] = 0.0f;
    #pragma unroll
    for (int t = 1; t <= 16; ++t) {
        pr[t] = pr[t - 1] * adr - pi[t - 1] * adi;
        pi[t] = pr[t - 1] * adi + pi[t - 1] * adr;
    }

    float* wsd = ws + (size_t)d * PD;
    __shared__ float sd[16 * NS];
    #pragma unroll
    for (int tau = 0; tau < 16; ++tau)
        sd[tau * NS + n] = (Cr * pr[tau] - Ci * pi[tau]) * Bd;
    __syncthreads();
    if (n < 16) {
        float s = 0.0f;
        for (int j = 0; j < NS; ++j) s += sd[n * NS + j];
        wsd[OFF_K + n] = s;                          // k_tau
    }

    // G: y_state[t] = Re(C * Ad^{t+1} * h) -> coef on (hr,hi); natural col order
    #pragma unroll
    for (int t = 0; t < 16; ++t) {
        wsd[OFF_G + t * 128 + 2 * n]     =  Cr * pr[t + 1] - Ci * pi[t + 1];
        wsd[OFF_G + t * 128 + 2 * n + 1] = -(Cr * pi[t + 1] + Ci * pr[t + 1]);
    }

    // W: h_new = Ad^16 h + sum_s Ad^{15-s} Bd u_s ; rows stored permuted
    const int M0 = permM(2 * n), M1 = permM(2 * n + 1);
    #pragma unroll
    for (int s = 0; s < 16; ++s) {
        wsd[OFF_W + M0 * 16 + s] = pr[15 - s] * Bd;  // -> hr
        wsd[OFF_W + M1 * 16 + s] = pi[15 - s] * Bd;  // -> hi
    }

    wsd[OFF_A16 + 2 * n]     = pr[16];
    wsd[OFF_A16 + 2 * n + 1] = pi[16];
}

// ---------------------------------------------------------------------------
// Kernel B: chunked scan. Block = 2 waves = channels (d0, d0+1); batch = WMMA
// N dim. Async global->LDS double buffering hides HBM/L2 latency; all global
// traffic is b64 (channel pairs), halving L2 request count.
// ---------------------------------------------------------------------------
__global__ void __launch_bounds__(64, 1)
s4_scan(const float* __restrict__ u, const float* __restrict__ Dv,
        const float* __restrict__ ws, float* __restrict__ y) {
    const int d0  = blockIdx.x * 2;
    const int tid = threadIdx.x;
    const int w   = tid >> 5;        // wave -> which channel of the pair
    const int l   = tid & 31;
    const int hl  = l >> 4;          // lane half
    const int b   = l & 15;          // batch col (B-frag) == matrix row t (A-frag)
    const int d   = d0 + w;
    const float* wsd = ws + (size_t)d * PD;

    // --- resident fragments (per wave / per channel) ---
    float kv[16];
    #pragma unroll
    for (int i = 0; i < 16; ++i) kv[i] = wsd[OFF_K + i];

    v2f kfrag[4];                                 // Toeplitz K (A-matrix)
    #pragma unroll
    for (int kk = 0; kk < 4; ++kk) {
        const int s0 = 4 * kk + 2 * hl;
        kfrag[kk].x = (b >= s0)     ? kv[b - s0]     : 0.0f;
        kfrag[kk].y = (b >= s0 + 1) ? kv[b - s0 - 1] : 0.0f;
    }
    v2f gfrag[32];                                // G (A-matrix, K-dim=128)
    #pragma unroll
    for (int kk = 0; kk < 32; ++kk) {
        const int c0 = 4 * kk + 2 * hl;
        gfrag[kk].x = wsd[OFF_G + b * 128 + c0];
        gfrag[kk].y = wsd[OFF_G + b * 128 + c0 + 1];
    }
    v2f wfrag[8][4];                              // W (8 row-tiles, A-matrix)
    #pragma unroll
    for (int r = 0; r < 8; ++r)
        #pragma unroll
        for (int kk = 0; kk < 4; ++kk) {
            const int s0 = 4 * kk + 2 * hl;
            wfrag[r][kk].x = wsd[OFF_W + (16 * r + b) * 16 + s0];
            wfrag[r][kk].y = wsd[OFF_W + (16 * r + b) * 16 + s0 + 1];
        }
    v2f a16[32], h[32];                           // state resident in B-frag form
    #pragma unroll
    for (int kk = 0; kk < 32; ++kk) {
        a16[kk].x = wsd[OFF_A16 + 4 * kk + 2 * hl];
        a16[kk].y = wsd[OFF_A16 + 4 * kk + 2 * hl + 1];
        h[kk].x = 0.0f; h[kk].y = 0.0f;
    }
    const float dskip = Dv[d];

    // LDS: two async input buffers + one output staging tile
    __shared__ float Ub[2][T * TP * 2];
    __shared__ float Yt[T * TP * 2];

    // async copy ownership: thread moves 4 (b,t) pairs, each 8B = (d0,d0+1)
    const int bb = tid >> 2;                       // 0..15
    uint64_t gaddr[4];
    uint32_t ldst0[4], ldst1[4];
    #pragma unroll
    for (int i = 0; i < 4; ++i) {
        const int tt = (tid & 3) * 4 + i;
        gaddr[i] = (uint64_t)(uintptr_t)(u + ((size_t)bb * LL + tt) * DM + d0);
        ldst0[i] = lds_off(&Ub[0][(tt * TP + bb) * 2]);
        ldst1[i] = lds_off(&Ub[1][(tt * TP + bb) * 2]);
    }
    // prologue: fetch chunk 0 into buffer 0
    #pragma unroll
    for (int i = 0; i < 4; ++i)
        asm volatile("global_load_async_to_lds_b64 %0, %1, off"
                     :: "v"(ldst0[i]), "v"(gaddr[i]) : "memory");

    for (int c = 0; c < NCHUNK; ++c) {
        const int l0 = c * T;
        // buffer for chunk c is ready once our own asyncs drained + barrier
        asm volatile("s_wait_asynccnt 0x0" ::: "memory");
        __syncthreads();
        const float* Ut = Ub[c & 1];

        // kick off chunk c+1 into the other buffer
        if (c + 1 < NCHUNK) {
            #pragma unroll
            for (int i = 0; i < 4; ++i) {
                gaddr[i] += (uint64_t)T * DM * sizeof(float);
                const uint32_t ldst = ((c + 1) & 1) ? ldst1[i] : ldst0[i];
                asm volatile("global_load_async_to_lds_b64 %0, %1, off"
                             :: "v"(ldst), "v"(gaddr[i]) : "memory");
            }
        }

        v2f ufrag[4];                             // U (B-matrix) for this channel
        #pragma unroll
        for (int kk = 0; kk < 4; ++kk) {
            const int s0 = 4 * kk + 2 * hl;
            ufrag[kk].x = Ut[(s0 * TP + b) * 2 + w];
            ufrag[kk].y = Ut[((s0 + 1) * TP + b) * 2 + w];
        }

        // y = K@U  +  G@H   (C/D layout: t rows, b cols)
        v8f acc = {};
        #pragma unroll
        for (int kk = 0; kk < 4; ++kk)
            acc = __builtin_amdgcn_wmma_f32_16x16x4_f32(
                false, kfrag[kk], false, ufrag[kk], (short)0, acc, false, false);
        #pragma unroll
        for (int kk = 0; kk < 32; ++kk)
            acc = __builtin_amdgcn_wmma_f32_16x16x4_f32(
                false, gfrag[kk], false, h[kk], (short)0, acc, false, false);

        // state carry: H = A16 (.) H + W@U  (permuted rows -> direct slot match)
        #pragma unroll
        for (int r = 0; r < 8; ++r) {
            v8f cw = {};
            #pragma unroll
            for (int kk = 0; kk < 4; ++kk)
                cw = __builtin_amdgcn_wmma_f32_16x16x4_f32(
                    false, wfrag[r][kk], false, ufrag[kk], (short)0, cw, false, false);
            #pragma unroll
            for (int p = 0; p < 4; ++p) {
                const int kk = 4 * r + p;
                v2f hn;
                hn.x = cw[2 * p]     + a16[kk].x * h[kk].x - a16[kk].y * h[kk].y;
                hn.y = cw[2 * p + 1] + a16[kk].y * h[kk].x + a16[kk].x * h[kk].y;
                h[kk] = hn;
            }
        }

        // skip connection, stage into LDS so stores can pair (d0,d0+1) as b64
        #pragma unroll
        for (int m = 0; m < 8; ++m) {
            const int t = m + 8 * hl;
            Yt[(t * TP + b) * 2 + w] = acc[m] + dskip * Ut[(t * TP + b) * 2 + w];
        }
        __syncthreads();

        #pragma unroll
        for (int i = 0; i < 4; ++i) {
            const int tt = (tid & 3) * 4 + i;
            const v2f val = *(const v2f*)&Yt[(tt * TP + bb) * 2];
            *(v2f*)(y + ((size_t)bb * LL + l0 + tt) * DM + d0) = val;
        }
        // next iteration's top barrier separates these Yt reads from rewrites
    }
}

// ---------------------------------------------------------------------------
// Kernel C: in-place LayerNorm over D=512 per (b,l) row.
// ---------------------------------------------------------------------------
__global__ void s4_ln(float* __restrict__ y, const float* __restrict__ w,
                      const float* __restrict__ bias) {
    float* p = y + (size_t)blockIdx.x * DM;
    const int t = threadIdx.x;                    // 256 threads, 2 elems each
    const float x0 = p[t], x1 = p[t + 256];
    __shared__ float s1[256], s2[256];
    s1[t] = x0 + x1;
    s2[t] = x0 * x0 + x1 * x1;
    __syncthreads();
    for (int off = 128; off > 0; off >>= 1) {
        if (t < off) { s1[t] += s1[t + off]; s2[t] += s2[t + off]; }
        __syncthreads();
    }
    const float mean = s1[0] * (1.0f / DM);
    const float var  = s2[0] * (1.0f / DM) - mean * mean;
    const float inv  = rsqrtf(var + 1e-5f);
    p[t]       = (x0 - mean) * inv * w[t]       + bias[t];
    p[t + 256] = (x1 - mean) * inv * w[t + 256] + bias[t + 256];
}

// ---------------------------------------------------------------------------
extern "C" void kernel_launch(void* const* d_in, const int* in_sizes, int n_in,
                              void* d_out, int out_size, void* d_ws, size_t ws_size,
                              hipStream_t stream) {
    const float* u   = (const float*)d_in[0];
    const float* lar = (const float*)d_in[1];
    const float* aim = (const float*)d_in[2];
    const float* Bm  = (const float*)d_in[3];
    const float* Cm  = (const float*)d_in[4];
    const float* Dv  = (const float*)d_in[5];
    const float* ldt = (const float*)d_in[6];
    const float* lnw = (const float*)d_in[7];
    const float* lnb = (const float*)d_in[8];
    float* out = (float*)d_out;
    float* ws  = (float*)d_ws;

    s4_tables<<<DM, NS, 0, stream>>>(lar, aim, Bm, Cm, ldt, ws);
    s4_scan<<<DM / 2, 64, 0, stream>>>(u, Dv, ws, out);
    s4_ln<<<NB * LL, 256, 0, stream>>>(out, lnw, lnb);
}